// KINET_DSMC_592705487079
// MI455X (gfx1250) — compile-verified
//
#include <hip/hip_runtime.h>
#include <hip/hip_bf16.h>

// KINET DSMC collision step, restructured for CDNA5 (gfx1250):
//   pairwise norms -> Gram matrices via V_WMMA_F32_16X16X4_F32 (K=16 = 4 chained K=4 WMMAs)
//   rand_vec stream (134MB) dominates -> memory bound ~11us; everything else fused/recomputed.

typedef __attribute__((ext_vector_type(2))) float v2f;
typedef __attribute__((ext_vector_type(8))) float v8f;

#define CHNL 16
#define NPART 1024

// ---------------------------------------------------------------- K1: prep
// vv = v + a*DT (DT=1); per-particle squared norms sx, sv; init v_r_max accumulator.
__global__ __launch_bounds__(256) void KINET_prep(
    const float* __restrict__ x, const float* __restrict__ v,
    const float* __restrict__ a, float* __restrict__ vvws,
    float* __restrict__ sxws, float* __restrict__ svws,
    unsigned* __restrict__ vrmax, int n) {
  int t = blockIdx.x * blockDim.x + threadIdx.x;
  int b = t / n, i = t % n;
  float sx = 0.f, sv = 0.f;
#pragma unroll
  for (int c = 0; c < CHNL; ++c) {
    size_t idx = ((size_t)b * CHNL + c) * n + i;
    float xx = x[idx];
    float vx = v[idx] + a[idx];          // DT = 1
    vvws[idx] = vx;
    sx = fmaf(xx, xx, sx);
    sv = fmaf(vx, vx, sv);
  }
  sxws[(size_t)b * n + i] = sx;
  svws[(size_t)b * n + i] = sv;
  if (i == 0) vrmax[b] = 0u;             // f32>=0: uint order == float order
}

// ---------------------------------------------------------------- K2: v_r_max
// One wave per (batch, j-tile of 16). B operand hoisted; loop i-tiles, Gram via WMMA,
// track max of vr^2 = sv_i + sv_j - 2*G, reduce, atomicMax.
__global__ __launch_bounds__(32) void KINET_vrmax(
    const float* __restrict__ vv, const float* __restrict__ sv,
    unsigned* __restrict__ vrmax, int n) {
  const int jt = n >> 4;
  const int b  = blockIdx.x / jt;
  const int j0 = (blockIdx.x % jt) << 4;
  const int l  = threadIdx.x;
  const int jl = l & 15, hf = l >> 4;    // hf: half-wave (lanes 16-31 hold K+... / M+8)
  const float* V   = vv + (size_t)b * CHNL * n;
  const float* svb = sv + (size_t)b * n;

  v2f Bv[4];                             // B 4x16 f32: VGPR0={K0|K2}, VGPR1={K1|K3}
#pragma unroll
  for (int t = 0; t < 4; ++t) {
    int kk = 4 * t + (hf << 1);
    Bv[t].x = V[(size_t)kk * n + j0 + jl];
    Bv[t].y = V[(size_t)(kk + 1) * n + j0 + jl];
  }
  const float svj = svb[j0 + jl];
  float mx = 0.f;

  for (int it = 0; it < (n >> 4); ++it) {
    const int i0 = it << 4;
    v2f Av[4];                           // A 16x4 f32: same lane striping, M = l&15
#pragma unroll
    for (int t = 0; t < 4; ++t) {
      int kk = 4 * t + (hf << 1);
      Av[t].x = V[(size_t)kk * n + i0 + jl];
      Av[t].y = V[(size_t)(kk + 1) * n + i0 + jl];
    }
    v8f cv = {};
#pragma unroll
    for (int t = 0; t < 4; ++t)
      cv = __builtin_amdgcn_wmma_f32_16x16x4_f32(false, Av[t], false, Bv[t],
                                                 (short)0, cv, false, false);
#pragma unroll
    for (int r = 0; r < 8; ++r) {        // C: VGPR r -> M = r (+8 for lanes 16-31)
      int i = i0 + r + hf * 8;
      float vr2 = svb[i] + svj - 2.f * cv[r];
      mx = fmaxf(mx, vr2);
    }
  }
#pragma unroll
  for (int off = 16; off; off >>= 1)
    mx = fmaxf(mx, __shfl_xor(mx, off, 32));
  if (l == 0) atomicMax(vrmax + b, __float_as_uint(mx));
}

// ---------------------------------------------------------------- K3: main
// One wave per (batch, j-tile). Recompute Gx/Gv tiles with WMMA, build collision mask,
// stream rand_vec only for colliding pairs, accumulate per-(c,j) update in registers.
__global__ __launch_bounds__(32) void KINET_collide(
    const float* __restrict__ x, const float* __restrict__ vv,
    const float* __restrict__ sx, const float* __restrict__ sv,
    const unsigned* __restrict__ vrmax, const float* __restrict__ rv,
    float* __restrict__ xout, float* __restrict__ vout,
    int n, long long mm) {
  __shared__ float vt[CHNL][16];         // vv[:, i-tile], filled from A registers

  const int jt = n >> 4;
  const int b  = blockIdx.x / jt;
  const int j0 = (blockIdx.x % jt) << 4;
  const int l  = threadIdx.x;
  const int jl = l & 15, hf = l >> 4;
  const int j  = j0 + jl;

  const float* X   = x  + (size_t)b * CHNL * n;
  const float* V   = vv + (size_t)b * CHNL * n;
  const float* sxb = sx + (size_t)b * n;
  const float* svb = sv + (size_t)b * n;
  const float* rvb = rv + (size_t)b * CHNL * mm;
  const float inv_vrmax = 1.0f / sqrtf(__uint_as_float(vrmax[b]));

  v2f Bx[4], Bv[4];                      // j-side operands: loop invariant
#pragma unroll
  for (int t = 0; t < 4; ++t) {
    int kk = 4 * t + (hf << 1);
    Bx[t].x = X[(size_t)kk * n + j];  Bx[t].y = X[(size_t)(kk + 1) * n + j];
    Bv[t].x = V[(size_t)kk * n + j];  Bv[t].y = V[(size_t)(kk + 1) * n + j];
  }
  const float sxj = sxb[j], svj = svb[j];

  float cnt = 0.f;
  float T[CHNL];
#pragma unroll
  for (int c = 0; c < CHNL; ++c) T[c] = 0.f;

  for (int it = 0; it < (n >> 4); ++it) {
    const int i0 = it << 4;
    v2f Ax[4], Av[4];
#pragma unroll
    for (int t = 0; t < 4; ++t) {
      int kk = 4 * t + (hf << 1);
      Ax[t].x = X[(size_t)kk * n + i0 + jl]; Ax[t].y = X[(size_t)(kk + 1) * n + i0 + jl];
      Av[t].x = V[(size_t)kk * n + i0 + jl]; Av[t].y = V[(size_t)(kk + 1) * n + i0 + jl];
    }
    __syncthreads();                     // prior-iter vt readers done
#pragma unroll
    for (int t = 0; t < 4; ++t) {        // stage vv i-tile to LDS from A regs (free)
      int kk = 4 * t + (hf << 1);
      vt[kk][jl]     = Av[t].x;
      vt[kk + 1][jl] = Av[t].y;
    }
    __syncthreads();

    v8f cx = {}, cv = {};
#pragma unroll
    for (int t = 0; t < 4; ++t) {
      cx = __builtin_amdgcn_wmma_f32_16x16x4_f32(false, Ax[t], false, Bx[t],
                                                 (short)0, cx, false, false);
      cv = __builtin_amdgcn_wmma_f32_16x16x4_f32(false, Av[t], false, Bv[t],
                                                 (short)0, cv, false, false);
    }

#pragma unroll
    for (int r = 0; r < 8; ++r) {
      const int il = r + hf * 8;         // local row in tile
      const int i  = i0 + il;
      float xr2 = sxb[i] + sxj - 2.f * cx[r];
      float vr2 = svb[i] + svj - 2.f * cv[r];
      float xr  = sqrtf(fmaxf(xr2, 0.f));
      float vr  = sqrtf(fmaxf(vr2, 0.f));
      float mask = vr * inv_vrmax * __expf(-xr);
      if (mask > 0.5f) {                 // diagonal self-excludes (vr==0)
        cnt += 1.f;
        int lo = min(i, j), hi_i = max(i, j);
        float svr = (i < j) ? vr : -vr;  // antisymmetric projection sign
        long long p = (long long)lo * n - (((long long)lo * (lo + 1)) >> 1)
                      + (hi_i - lo - 1); // triu_indices(n, k=1) linear index
        const float* rvp = rvb + p;
        __builtin_prefetch(rvp, 0, 0);
#pragma unroll
        for (int c = 0; c < CHNL; ++c)
          T[c] += 0.5f * vt[c][il] - svr * rvp[(size_t)c * mm];
      }
    }
  }

  // combine the two half-wave partners that own the same column j
  cnt += __shfl_xor(cnt, 16, 32);
#pragma unroll
  for (int c = 0; c < CHNL; ++c) T[c] += __shfl_xor(T[c], 16, 32);

  if (l < 16) {                          // lanes 0-15 write the 16 columns
#pragma unroll
    for (int c = 0; c < CHNL; ++c) {
      size_t idx = ((size_t)b * CHNL + c) * n + j;
      float vvjc = V[(size_t)c * n + j];
      float nv = vvjc + 0.5f * cnt * vvjc - T[c];   // vv + sum_i cmask*delta_v
      vout[idx] = nv;
      xout[idx] = X[(size_t)c * n + j] + nv;        // x + vv_new*DT (DT=1)
    }
  }
}

// ---------------------------------------------------------------- launch
extern "C" void kernel_launch(void* const* d_in, const int* in_sizes, int n_in,
                              void* d_out, int out_size, void* d_ws, size_t ws_size,
                              hipStream_t stream) {
  const float* x  = (const float*)d_in[0];
  const float* v  = (const float*)d_in[1];
  const float* a  = (const float*)d_in[2];
  const float* rv = (const float*)d_in[3];

  const int n  = NPART;
  const int bs = in_sizes[0] / (CHNL * n);
  const long long mm = (long long)n * (n - 1) / 2;

  float* ws    = (float*)d_ws;
  float* vvws  = ws;                                     // bs*16*n
  float* sxws  = vvws + (size_t)bs * CHNL * n;           // bs*n
  float* svws  = sxws + (size_t)bs * n;                  // bs*n
  unsigned* vrmax = (unsigned*)(svws + (size_t)bs * n);  // bs

  float* xout = (float*)d_out;
  float* vout = xout + (size_t)bs * CHNL * n;

  KINET_prep<<<(bs * n) / 256, 256, 0, stream>>>(x, v, a, vvws, sxws, svws, vrmax, n);
  KINET_vrmax<<<bs * (n >> 4), 32, 0, stream>>>(vvws, svws, vrmax, n);
  KINET_collide<<<bs * (n >> 4), 32, 0, stream>>>(x, vvws, sxws, svws, vrmax, rv,
                                                  xout, vout, n, mm);
}